// CPTensorProduct_65876208386288
// MI455X (gfx1250) — compile-verified
//
#include <hip/hip_runtime.h>

typedef __attribute__((ext_vector_type(2))) float v2f;
typedef __attribute__((ext_vector_type(4))) float v4f;
typedef __attribute__((ext_vector_type(8))) float v8f;

#define LDA  64
#define LDB  64
#define LDC2 68   // padded: strided lane access, 68*ln mod 64 distinct, 16B-aligned rows
#define LDX1 64
#define LDX2 32
#define LDW  36   // padded: 36*ln mod 64 = 4*(9*ln mod 16) distinct; rows 144B (16B aligned)
#define LDT2 36
#define LDP  72   // padded: 2*72 mod 64 = 16 -> upper/lower lane halves hit disjoint banks

__device__ __forceinline__ v8f wmma4(v2f a, v2f b, v8f c) {
  // D = A(16x4,f32) * B(4x16,f32) + C(16x16,f32)
  return __builtin_amdgcn_wmma_f32_16x16x4_f32(false, a, false, b, (short)0, c,
                                               false, false);
}

__global__ __launch_bounds__(128, 1)
void cp_tp_kernel(const float* __restrict__ x1, const float* __restrict__ x2,
                  const float* __restrict__ w,  const float* __restrict__ Ag,
                  const float* __restrict__ Bg, const float* __restrict__ Cg,
                  float* __restrict__ out)
{
  __shared__ __align__(16) float sA [16 * LDA];   // A[i][r]
  __shared__ __align__(16) float sB [16 * LDB];   // B[j][r]
  __shared__ __align__(16) float sC [16 * LDC2];  // C[c][r]
  __shared__ __align__(16) float sX1[16 * LDX1];  // x1[i][o]
  __shared__ __align__(16) float sX2[16 * LDX2];  // x2[j][v]
  __shared__ __align__(16) float sW [64 * LDW];   // w[o][v]
  __shared__ __align__(16) float sT2[64 * LDT2];  // t2[r][v]
  __shared__ __align__(16) float sP [64 * LDP];   // (t1*t3)[r][o]

  const int tid  = threadIdx.x;
  const int lane = tid & 31;
  const int wv   = tid >> 5;     // wave id 0..3 -> rank strip
  const int kh   = lane >> 4;    // lane half: selects K pair {0,1} vs {2,3}
  const int ln   = lane & 15;
  const long b   = blockIdx.x;

  // ---------------- cooperative LDS staging ----------------
  const v4f* gx1 = (const v4f*)(x1 + b * 1024);
  const v4f* gx2 = (const v4f*)(x2 + b * 512);
  const v4f* gw  = (const v4f*)(w  + b * 2048);
  const v4f* gA  = (const v4f*)Ag;
  const v4f* gB  = (const v4f*)Bg;
  const v4f* gC  = (const v4f*)Cg;

#pragma unroll
  for (int q = 0; q < 2; ++q) {             // A, B, x1: 256 v4f each (flat)
    int i = tid + q * 128;
    *(v4f*)&sA [i * 4] = gA[i];
    *(v4f*)&sB [i * 4] = gB[i];
    *(v4f*)&sX1[i * 4] = __builtin_nontemporal_load(&gx1[i]);
  }
  *(v4f*)&sX2[tid * 4] = __builtin_nontemporal_load(&gx2[tid]);  // 128 v4f
#pragma unroll
  for (int q = 0; q < 2; ++q) {             // C: 256 v4f, row stride 68
    int i = tid + q * 128;
    int r = i >> 4, c4 = i & 15;
    *(v4f*)&sC[r * LDC2 + c4 * 4] = gC[i];
  }
#pragma unroll
  for (int q = 0; q < 4; ++q) {             // w: 512 v4f, row stride 36
    int i = tid + q * 128;
    int r = i >> 3, c4 = i & 7;
    *(v4f*)&sW[r * LDW + c4 * 4] = __builtin_nontemporal_load(&gw[i]);
  }
  __syncthreads();

  // ---------------- phase 1: per-wave rank strip r in [16*wv, 16*wv+16) ---
  v8f t1acc[4], t3acc[4];

  // t1[r][o] = sum_i A[i][r] * x1[i][o]   (M=16 strip, N=64, K=16)
#pragma unroll
  for (int nt = 0; nt < 4; ++nt) {
    v8f acc = {0.f,0.f,0.f,0.f,0.f,0.f,0.f,0.f};
#pragma unroll
    for (int ks = 0; ks < 4; ++ks) {
      int k = 4 * ks + 2 * kh;
      v2f a, bb;
      a.x  = sA [ k      * LDA  + 16 * wv + ln];
      a.y  = sA [(k + 1) * LDA  + 16 * wv + ln];
      bb.x = sX1[ k      * LDX1 + 16 * nt + ln];
      bb.y = sX1[(k + 1) * LDX1 + 16 * nt + ln];
      acc = wmma4(a, bb, acc);
    }
    t1acc[nt] = acc;
  }

  // t2[r][v] = sum_j B[j][r] * x2[j][v]   (M=16 strip, N=32, K=16)
  v8f t2acc[2];
#pragma unroll
  for (int nt = 0; nt < 2; ++nt) {
    v8f acc = {0.f,0.f,0.f,0.f,0.f,0.f,0.f,0.f};
#pragma unroll
    for (int ks = 0; ks < 4; ++ks) {
      int k = 4 * ks + 2 * kh;
      v2f a, bb;
      a.x  = sB [ k      * LDB  + 16 * wv + ln];
      a.y  = sB [(k + 1) * LDB  + 16 * wv + ln];
      bb.x = sX2[ k      * LDX2 + 16 * nt + ln];
      bb.y = sX2[(k + 1) * LDX2 + 16 * nt + ln];
      acc = wmma4(a, bb, acc);
    }
    t2acc[nt] = acc;
  }
  // spill t2 strip to LDS (C-layout -> row-major); same-wave DS ops are in-order
#pragma unroll
  for (int nt = 0; nt < 2; ++nt)
#pragma unroll
    for (int g = 0; g < 8; ++g)
      sT2[(16 * wv + kh * 8 + g) * LDT2 + 16 * nt + ln] = t2acc[nt][g];

  // t3[r][o] = sum_v t2[r][v] * w[o][v]   (M=16 strip, N=64, K=32)
#pragma unroll
  for (int nt = 0; nt < 4; ++nt) {
    v8f acc = {0.f,0.f,0.f,0.f,0.f,0.f,0.f,0.f};
#pragma unroll
    for (int kv = 0; kv < 8; ++kv) {
      int k = 4 * kv + 2 * kh;
      v2f a, bb;
      a.x  = sT2[(16 * wv + ln) * LDT2 + k];       // A read as row-major M x K
      a.y  = sT2[(16 * wv + ln) * LDT2 + k + 1];
      bb.x = sW [(16 * nt + ln) * LDW  + k];       // B = w^T, read transposed
      bb.y = sW [(16 * nt + ln) * LDW  + k + 1];
      acc = wmma4(a, bb, acc);
    }
    t3acc[nt] = acc;
  }

  // P = t1 .* t3  (both in identical C/D layout) -> shared for final contraction
#pragma unroll
  for (int nt = 0; nt < 4; ++nt)
#pragma unroll
    for (int g = 0; g < 8; ++g)
      sP[(16 * wv + kh * 8 + g) * LDP + 16 * nt + ln] = t1acc[nt][g] * t3acc[nt][g];

  __syncthreads();

  // ---------------- phase 2: out[c][o] = sum_r C[c][r] * P[r][o] ----------
  // wave wv computes the 16x16 output tile for o in [16*wv, 16*wv+16), K=64
  v8f oacc = {0.f,0.f,0.f,0.f,0.f,0.f,0.f,0.f};
#pragma unroll
  for (int ks = 0; ks < 16; ++ks) {
    int k = 4 * ks + 2 * kh;
    v2f a, bb;
    a.x  = sC[ln * LDC2 + k];                      // A = C, row-major 16 x 64
    a.y  = sC[ln * LDC2 + k + 1];
    bb.x = sP[ k      * LDP + 16 * wv + ln];       // B = P, row-major K x N
    bb.y = sP[(k + 1) * LDP + 16 * wv + ln];
    oacc = wmma4(a, bb, oacc);
  }

  // store out[b][c][o]: row c = kh*8+g, col o = 16*wv+ln
  float* og = out + b * 1024 + (kh * 8) * 64 + 16 * wv + ln;
#pragma unroll
  for (int g = 0; g < 8; ++g)
    __builtin_nontemporal_store(oacc[g], &og[g * 64]);
}

extern "C" void kernel_launch(void* const* d_in, const int* in_sizes, int n_in,
                              void* d_out, int out_size, void* d_ws, size_t ws_size,
                              hipStream_t stream) {
  const float* x1 = (const float*)d_in[0];
  const float* x2 = (const float*)d_in[1];
  const float* w  = (const float*)d_in[2];
  const float* A  = (const float*)d_in[3];
  const float* B  = (const float*)d_in[4];
  const float* C  = (const float*)d_in[5];
  float* out = (float*)d_out;

  const int batch = in_sizes[0] / (16 * 64);   // 32768
  cp_tp_kernel<<<batch, 128, 0, stream>>>(x1, x2, w, A, B, C, out);
}